// GraphAttentionWithRelationLayer_1047972020769
// MI455X (gfx1250) — compile-verified
//
#include <hip/hip_runtime.h>
#include <math.h>

typedef float v2f __attribute__((ext_vector_type(2)));
typedef float v4f __attribute__((ext_vector_type(4)));
typedef float v8f __attribute__((ext_vector_type(8)));

#define NN    1024
#define F_IN  128
#define F_OUT 128
#define LRELU_ALPHA 0.2f
#define NEG_BIG (-9.0e15f)

// ---------------------------------------------------------------------------
// Kernel 1: Wh[1024,128] = h[1024,128] @ W_w[128,128] + W_b
// One wave (32 threads) per 16x16 output tile, K-loop of 4 per WMMA issue.
// A-frag layout (ISA 7.12.2, 32-bit A 16x4): lanes 0-15 hold K=k0,k0+1 for
// M=lane; lanes 16-31 hold K=k0+2,k0+3. B-frag mirrors C/D striping.
// ---------------------------------------------------------------------------
__global__ void gemm_wh_kernel(const float* __restrict__ h,
                               const float* __restrict__ Ww,
                               const float* __restrict__ Wb,
                               float* __restrict__ Wh) {
    const int lane = threadIdx.x;        // 0..31
    const int half = lane >> 4;          // 0 or 1
    const int lm   = lane & 15;
    const int tn   = blockIdx.x * 16;    // output col tile
    const int tm   = blockIdx.y * 16;    // output row tile

    v8f c = {};
    const float* hrow = h + (size_t)(tm + lm) * F_IN;

    for (int k0 = 0; k0 < F_IN; k0 += 4) {
        const int ka = k0 + 2 * half;
        v2f a, b;
        a.x = hrow[ka];
        a.y = hrow[ka + 1];
        b.x = Ww[(size_t)ka * F_OUT + tn + lm];
        b.y = Ww[(size_t)(ka + 1) * F_OUT + tn + lm];
        c = __builtin_amdgcn_wmma_f32_16x16x4_f32(false, a, false, b,
                                                  (short)0, c, false, false);
    }

    const float bias = Wb[tn + lm];
#pragma unroll
    for (int r = 0; r < 8; ++r) {
        const int row = tm + r + 8 * half;
        Wh[(size_t)row * F_OUT + tn + lm] = c[r] + bias;
    }
}

// ---------------------------------------------------------------------------
// Kernel 2: s_i[r] = Wh[r,:] . a_i ; s_j[r] = Wh[r,:] . a_j
// One wave per row; each lane owns 4 consecutive floats (b128 load).
// ---------------------------------------------------------------------------
__global__ void scores_kernel(const float* __restrict__ Wh,
                              const float* __restrict__ aw,
                              float* __restrict__ s_i,
                              float* __restrict__ s_j) {
    const int wave = threadIdx.x >> 5;
    const int lane = threadIdx.x & 31;
    const int row  = blockIdx.x * (blockDim.x >> 5) + wave;

    const v4f w  = ((const v4f*)(Wh + (size_t)row * F_OUT))[lane];
    const v4f ai = ((const v4f*)(aw))[lane];
    const v4f aj = ((const v4f*)(aw + F_OUT))[lane];

    float si = w.x * ai.x + w.y * ai.y + w.z * ai.z + w.w * ai.w;
    float sj = w.x * aj.x + w.y * aj.y + w.z * aj.z + w.w * aj.w;
#pragma unroll
    for (int m = 16; m >= 1; m >>= 1) {
        si += __shfl_xor(si, m, 32);
        sj += __shfl_xor(sj, m, 32);
    }
    if (lane == 0) { s_i[row] = si; s_j[row] = sj; }
}

// ---------------------------------------------------------------------------
// Kernel 3 (HBM-bound, ~1 GB streamed): per (i,j) dot 128-float rows of e_h
// and dialogue against register-resident a_e / a_d; fuse +s_i+s_j+a_b,
// leaky-relu, adjacency mask. One block per i (8 waves), wave per j row.
// v4f-per-lane = 512B fully-coalesced global_load_b128 per wave per row.
// Streaming tensors are read-once -> non-temporal loads (TH=NT) so the 1 GB
// sweep does not evict the L2-resident GEMM operands (Wh, P).
// ---------------------------------------------------------------------------
__global__ void escore_kernel(const float* __restrict__ e_h,
                              const float* __restrict__ de,
                              const int*   __restrict__ adj,
                              const float* __restrict__ s_i,
                              const float* __restrict__ s_j,
                              const float* __restrict__ aw,
                              const float* __restrict__ ab,
                              float* __restrict__ e) {
    const int i      = blockIdx.x;
    const int wave   = threadIdx.x >> 5;
    const int lane   = threadIdx.x & 31;
    const int nwaves = blockDim.x >> 5;

    const v4f ae = ((const v4f*)(aw + 2 * F_OUT))[lane];
    const v4f ad = ((const v4f*)(aw + 2 * F_OUT + F_IN))[lane];
    const float si   = s_i[i];
    const float bias = ab[0];

    for (int j = wave; j < NN; j += nwaves) {
        const size_t off = ((size_t)i * NN + j) * F_IN;
        const v4f ev = __builtin_nontemporal_load((const v4f*)(e_h + off) + lane);
        const v4f dv = __builtin_nontemporal_load((const v4f*)(de + off) + lane);

        // prefetch the next rows this wave will touch (global_prefetch_b8)
        __builtin_prefetch(e_h + off + (size_t)nwaves * F_IN + 4 * lane, 0, 1);
        __builtin_prefetch(de  + off + (size_t)nwaves * F_IN + 4 * lane, 0, 1);

        float acc = ev.x * ae.x + ev.y * ae.y + ev.z * ae.z + ev.w * ae.w
                  + dv.x * ad.x + dv.y * ad.y + dv.z * ad.z + dv.w * ad.w;
#pragma unroll
        for (int m = 16; m >= 1; m >>= 1) acc += __shfl_xor(acc, m, 32);

        if (lane == 0) {
            const int a = __builtin_nontemporal_load(adj + (size_t)i * NN + j);
            float v = si + s_j[j] + acc + bias;
            v = v > 0.0f ? v : LRELU_ALPHA * v;  // leaky relu
            if (a <= 0) v = NEG_BIG;             // adjacency mask
            e[(size_t)i * NN + j] = v;
        }
    }
}

// ---------------------------------------------------------------------------
// Kernel 4: row-wise softmax over e[i, 0..1023], in place.
// ---------------------------------------------------------------------------
__global__ void softmax_kernel(float* __restrict__ e) {
    __shared__ float red[32];
    const int i    = blockIdx.x;
    float* row     = e + (size_t)i * NN;
    const int t    = threadIdx.x;
    const int wave = t >> 5;
    const int lane = t & 31;
    const int nwaves = blockDim.x >> 5;

    // ---- row max ----
    float m = -INFINITY;
    for (int j = t; j < NN; j += blockDim.x) m = fmaxf(m, row[j]);
#pragma unroll
    for (int s = 16; s >= 1; s >>= 1) m = fmaxf(m, __shfl_xor(m, s, 32));
    if (lane == 0) red[wave] = m;
    __syncthreads();
    m = (lane < nwaves) ? red[lane] : -INFINITY;
#pragma unroll
    for (int s = 16; s >= 1; s >>= 1) m = fmaxf(m, __shfl_xor(m, s, 32));
    __syncthreads();   // all reads of red done before reuse

    // ---- exp + sum ----
    float sum = 0.0f;
    for (int j = t; j < NN; j += blockDim.x) {
        const float ex = __expf(row[j] - m);
        row[j] = ex;
        sum += ex;
    }
#pragma unroll
    for (int s = 16; s >= 1; s >>= 1) sum += __shfl_xor(sum, s, 32);
    if (lane == 0) red[wave] = sum;
    __syncthreads();
    sum = (lane < nwaves) ? red[lane] : 0.0f;
#pragma unroll
    for (int s = 16; s >= 1; s >>= 1) sum += __shfl_xor(sum, s, 32);

    const float inv = 1.0f / sum;
    for (int j = t; j < NN; j += blockDim.x) row[j] *= inv;
}

// ---------------------------------------------------------------------------
// Kernel 5: out = elu(P[1024,1024] @ Wh[1024,128]) via f32 WMMA.
// Wave per 16x16 tile; P and Wh are L2-resident (NT loads above kept them so).
// ---------------------------------------------------------------------------
__global__ void gemm_out_kernel(const float* __restrict__ P,
                                const float* __restrict__ Wh,
                                float* __restrict__ out) {
    const int lane = threadIdx.x;
    const int half = lane >> 4;
    const int lm   = lane & 15;
    const int tn   = blockIdx.x * 16;
    const int tm   = blockIdx.y * 16;

    v8f c = {};
    const float* prow = P + (size_t)(tm + lm) * NN;

    for (int k0 = 0; k0 < NN; k0 += 4) {
        const int ka = k0 + 2 * half;
        v2f a, b;
        a.x = prow[ka];
        a.y = prow[ka + 1];
        b.x = Wh[(size_t)ka * F_OUT + tn + lm];
        b.y = Wh[(size_t)(ka + 1) * F_OUT + tn + lm];
        c = __builtin_amdgcn_wmma_f32_16x16x4_f32(false, a, false, b,
                                                  (short)0, c, false, false);
    }

#pragma unroll
    for (int r = 0; r < 8; ++r) {
        const int row = tm + r + 8 * half;
        const float x = c[r];
        out[(size_t)row * F_OUT + tn + lm] = (x > 0.0f) ? x : (__expf(x) - 1.0f); // elu
    }
}

// ---------------------------------------------------------------------------
extern "C" void kernel_launch(void* const* d_in, const int* in_sizes, int n_in,
                              void* d_out, int out_size, void* d_ws, size_t ws_size,
                              hipStream_t stream) {
    const float* h   = (const float*)d_in[0];   // (1024,128)
    const float* e_h = (const float*)d_in[1];   // (1024,1024,128)
    const float* de  = (const float*)d_in[2];   // (1024,1024,128)
    const int*   adj = (const int*)d_in[3];     // (1024,1024)
    const float* Ww  = (const float*)d_in[4];   // (128,128)
    const float* Wb  = (const float*)d_in[5];   // (128,)
    const float* aw  = (const float*)d_in[6];   // (512,1)
    const float* ab  = (const float*)d_in[7];   // (1,)
    float* out = (float*)d_out;                 // (1024,128)

    // workspace layout
    char*  ws  = (char*)d_ws;
    float* Wh  = (float*)ws;                          // 1024*128*4 = 512 KB
    float* s_i = (float*)(ws + 524288);               // 4 KB
    float* s_j = (float*)(ws + 524288 + 4096);        // 4 KB
    float* e   = (float*)(ws + 524288 + 8192);        // 1024*1024*4 = 4 MB

    // 1) Wh = h @ W_w + W_b   (WMMA f32)
    gemm_wh_kernel<<<dim3(F_OUT / 16, NN / 16), 32, 0, stream>>>(h, Ww, Wb, Wh);
    // 2) s_i, s_j
    scores_kernel<<<NN / 4, 128, 0, stream>>>(Wh, aw, s_i, s_j);
    // 3) e scores: leaky_relu(s_i+s_j+<e_h,a_e>+<de,a_d>+b) masked by adj
    escore_kernel<<<NN, 256, 0, stream>>>(e_h, de, adj, s_i, s_j, aw, ab, e);
    // 4) softmax rows (in place -> attention P)
    softmax_kernel<<<NN, 256, 0, stream>>>(e);
    // 5) out = elu(P @ Wh)   (WMMA f32)
    gemm_out_kernel<<<dim3(F_OUT / 16, NN / 16), 32, 0, stream>>>(e, Wh, out);
}